// KoleoLoss_70514773066231
// MI455X (gfx1250) — compile-verified
//
#include <hip/hip_runtime.h>
#include <math.h>

typedef float v2f __attribute__((ext_vector_type(2)));
typedef float v8f __attribute__((ext_vector_type(8)));

#define DIMS 64
#define ROWS_PER_BLOCK 128   // 8 waves x 16 rows
#define PANEL_COLS 128       // columns staged in LDS per iteration
#define LDS_STRIDE 68        // 64 + 4 pad: 16B-aligned b128 stores, conflict-free ds_load_b64
#define PANEL_FLOATS (PANEL_COLS * LDS_STRIDE)

// Issue this thread's share of one panel as async memory->LDS copies (b128 each:
// 16B per lane per instruction, 8 instructions per thread = 32KB per block).
// GVS addressing: saddr = uniform base, vaddr = per-lane 32-bit byte offset,
// vdst = per-lane LDS byte address (low 32 bits of the generic pointer, per the
// flat-aperture mapping).
__device__ __forceinline__ void stage_panel_async(const float* __restrict__ v,
                                                  int colBase, float* dstPanel,
                                                  int tid) {
#pragma unroll
  for (int i = 0; i < (PANEL_COLS * DIMS / 4) / 256; ++i) {   // 8 iterations
    int e  = tid + i * 256;          // float4 index in panel
    int c  = e >> 4;                 // column within panel
    int k4 = e & 15;                 // float4 within the 64-float row
    unsigned gofs = (unsigned)(((size_t)(colBase + c) * DIMS + k4 * 4) * sizeof(float));
    unsigned lofs = (unsigned)(size_t)(dstPanel + (c * LDS_STRIDE + k4 * 4));
    asm volatile("global_load_async_to_lds_b128 %0, %1, %2 offset:0"
                 :
                 : "v"(lofs), "v"(gofs), "s"(v)
                 : "memory");
  }
}

__device__ __forceinline__ void wait_async_all() {
  asm volatile("s_wait_asynccnt 0x0" ::: "memory");
}

// ---------------------------------------------------------------------------
// Kernel 1: fused Gram-matrix (fp32 WMMA) + per-row argmax (nearest neighbor).
// Never materializes the 16384^2 matrix; argmax lives in accumulator regs.
// Double-buffered async LDS panels overlap global traffic with WMMA compute.
// ---------------------------------------------------------------------------
__global__ __launch_bounds__(256)
void koleo_argmax_wmma(const float* __restrict__ v, int* __restrict__ nnIdx, int N) {
  __shared__ float panel[2][PANEL_FLOATS];

  const int tid  = threadIdx.x;
  const int lane = tid & 31;
  const int wave = tid >> 5;
  const int half = lane >> 4;   // 0: lanes 0-15, 1: lanes 16-31
  const int l15  = lane & 15;

  const int rowTile = blockIdx.x * ROWS_PER_BLOCK + wave * 16;

  // A fragments for this wave's 16 rows: 16 chunks of 16x4 fp32, resident for
  // the whole kernel. ISA layout: lane L<16 holds A[M=L][k0..k0+1]; lane L>=16
  // holds K=k0+2..k0+3.
  v2f a[16];
  {
    const float* ap = v + (size_t)(rowTile + l15) * DIMS + 2 * half;
#pragma unroll
    for (int kc = 0; kc < 16; ++kc)
      a[kc] = *(const v2f*)(ap + kc * 4);
  }

  float best[8];
  int   bidx[8];
#pragma unroll
  for (int r = 0; r < 8; ++r) { best[r] = -3.0e38f; bidx[r] = 0; }

  const int numPanels = N / PANEL_COLS;
  stage_panel_async(v, 0, panel[0], tid);            // prime buffer 0

  for (int p = 0; p < numPanels; ++p) {
    const int buf     = p & 1;
    const int colBase = p * PANEL_COLS;

    wait_async_all();        // this wave's async writes into panel[buf] done
    __syncthreads();         // ...and everyone else's too

    if (p + 1 < numPanels)   // kick off next panel into the other buffer
      stage_panel_async(v, colBase + PANEL_COLS, panel[buf ^ 1], tid);

#pragma unroll 1
    for (int ct = 0; ct < PANEL_COLS / 16; ++ct) {
      // Preload all 16 B fragments so the ds_load latency is paid once, then
      // run the 16-WMMA accumulation chain back-to-back.
      // B layout mirrors A: lane L holds B[k][n=L&15] for k=k0+2*(L>>4), +1;
      // B[k][n] = v[col_n][k] -> contiguous float2 per lane from LDS.
      const float* bp = &panel[buf][(ct * 16 + l15) * LDS_STRIDE + 2 * half];
      v2f b[16];
#pragma unroll
      for (int kc = 0; kc < 16; ++kc)
        b[kc] = *(const v2f*)(bp + kc * 4);

      v8f acc = {0.f, 0.f, 0.f, 0.f, 0.f, 0.f, 0.f, 0.f};
#pragma unroll
      for (int kc = 0; kc < 16; ++kc)
        acc = __builtin_amdgcn_wmma_f32_16x16x4_f32(
            /*neg_a=*/false, a[kc], /*neg_b=*/false, b[kc],
            /*c_mod=*/(short)0, acc, /*reuse_a=*/false, /*reuse_b=*/false);

      const int colTileBase = colBase + ct * 16;

      // Diagonal tile: both operands are wave-uniform, so this is a scalar
      // branch taken once per wave over the whole kernel.
      if (colTileBase == rowTile) {
        // C layout: VGPR r, lane L -> (M = r + 8*(L>>4), N = L&15).
        // col==row iff l15 == r + 8*half.
#pragma unroll
        for (int r = 0; r < 8; ++r)
          if (l15 == r + 8 * half) acc[r] = -1.0f;   // diag := -1, like reference
      }

      const int col = colTileBase + l15;
#pragma unroll
      for (int r = 0; r < 8; ++r)
        if (acc[r] > best[r]) { best[r] = acc[r]; bidx[r] = col; }  // strict > keeps first max

    }
    __syncthreads();         // done reading panel[buf] before p+2 overwrites it
  }

  // Cross-lane argmax within each 16-lane half (xor masks 1..8 never cross bit 4).
#pragma unroll
  for (int r = 0; r < 8; ++r) {
    float bv = best[r];
    int   bi = bidx[r];
#pragma unroll
    for (int m = 8; m >= 1; m >>= 1) {
      float ov = __shfl_xor(bv, m, 32);
      int   oi = __shfl_xor(bi, m, 32);
      if (ov > bv || (ov == bv && oi < bi)) { bv = ov; bi = oi; }  // lowest-index tie-break
    }
    if (l15 == 0) nnIdx[rowTile + r + 8 * half] = bi;
  }
}

// ---------------------------------------------------------------------------
// Kernel 2: per-row KoLeo term. One wave per row; eps added per component
// before the norm, matching torch.nn.PairwiseDistance semantics.
// ---------------------------------------------------------------------------
__global__ __launch_bounds__(256)
void koleo_rowloss(const float* __restrict__ v, const int* __restrict__ nnIdx,
                   float* __restrict__ rowLoss, int N) {
  const int row  = blockIdx.x * 8 + (threadIdx.x >> 5);
  const int lane = threadIdx.x & 31;
  const int nn   = nnIdx[row];

  const float* ap = v + (size_t)row * DIMS;
  const float* bp = v + (size_t)nn  * DIMS;
  float x0 = ap[lane]      - bp[lane]      + 1e-6f;
  float x1 = ap[lane + 32] - bp[lane + 32] + 1e-6f;
  float s  = x0 * x0 + x1 * x1;
#pragma unroll
  for (int m = 16; m >= 1; m >>= 1) s += __shfl_xor(s, m, 32);

  if (lane == 0) {
    float dist = sqrtf(s);
    float kl   = -logf(dist * (float)N);
    rowLoss[row] = (kl < 0.0f) ? 0.0f : kl;
  }
}

// ---------------------------------------------------------------------------
// Kernel 3: deterministic single-block mean reduction (no float atomics, so
// the result is bitwise identical on every replay).
// ---------------------------------------------------------------------------
__global__ __launch_bounds__(256)
void koleo_reduce(const float* __restrict__ rowLoss, float* __restrict__ out, int N) {
  __shared__ float sm[256];
  const int t = threadIdx.x;
  float s = 0.0f;
  for (int i = t; i < N; i += 256) s += rowLoss[i];   // fixed per-thread order
  sm[t] = s;
  __syncthreads();
  for (int ofs = 128; ofs >= 1; ofs >>= 1) {
    if (t < ofs) sm[t] += sm[t + ofs];
    __syncthreads();
  }
  if (t == 0) out[0] = sm[0] / (float)N;
}

extern "C" void kernel_launch(void* const* d_in, const int* in_sizes, int n_in,
                              void* d_out, int out_size, void* d_ws, size_t ws_size,
                              hipStream_t stream) {
  const float* v = (const float*)d_in[0];
  const int N = in_sizes[0] / DIMS;   // 16384

  int*   nnIdx   = (int*)d_ws;
  float* rowLoss = (float*)((char*)d_ws + (size_t)N * sizeof(int));

  koleo_argmax_wmma<<<N / ROWS_PER_BLOCK, 256, 0, stream>>>(v, nnIdx, N);
  koleo_rowloss<<<N / 8, 256, 0, stream>>>(v, nnIdx, rowLoss, N);
  koleo_reduce<<<1, 256, 0, stream>>>(rowLoss, (float*)d_out, N);
}